// Net_61057255080062
// MI455X (gfx1250) — compile-verified
//
#include <hip/hip_runtime.h>
#include <math.h>

// ---------------------------------------------------------------------------
// 2-layer GCN (1 -> 16 -> 10) on N=100000 nodes, E=3200000 edges, + log_softmax.
//
// Design (MI455X, wave32, 192MB L2, 23.3 TB/s):
//  * Problem is scatter/gather bound, not FLOP bound (~64 MFLOP total dense).
//  * Entire working set (~65MB) fits in L2 -> edge passes run at L2 bandwidth,
//    float atomics resolve in L2 atomic units.
//  * Layer 1 collapses to a per-node scalar (x is [N,1]):
//        s[i]   = sum_{(r,c=i)} dinv[r]*x[r]          (4B scatter per edge)
//        a[i]   = dinv[i]*s[i] + dinv[i]^2*x[i]
//        h[i,k] = relu(a[i]*W1[k] + b1[k])
//  * Layer 2: transform-then-aggregate, norm factored per-endpoint:
//        mvec[i,:] = dinv[i] * (h[i,:] @ W2)           (exact f32 WMMA 16x16x4)
//        t[c,:]    = sum_{(r,c)} mvec[r,:]             (40B gather+scatter per edge)
//        out[c,:]  = log_softmax(dinv[c]*t[c,:] + dinv[c]^2*(h[c]@W2) + b2)
// ---------------------------------------------------------------------------

typedef __attribute__((ext_vector_type(2))) float v2f;
typedef __attribute__((ext_vector_type(8))) float v8f;

#define NN 100000
#define NE 3200000
#define F2 10

__global__ void zero_kernel(float* __restrict__ p, int n) {
  int i = blockIdx.x * blockDim.x + threadIdx.x;
  if (i < n) p[i] = 0.0f;
}

__global__ void deg_kernel(const long long* __restrict__ col, float* __restrict__ deg) {
  int i = blockIdx.x * blockDim.x + threadIdx.x;
  if (i < NE) atomicAdd(&deg[(int)col[i]], 1.0f);
}

__global__ void dinv_kernel(const float* __restrict__ deg, float* __restrict__ dinv) {
  int i = blockIdx.x * blockDim.x + threadIdx.x;
  if (i < NN) dinv[i] = rsqrtf(deg[i] + 1.0f);  // +1 = self loop; always > 0
}

__global__ void scatter1_kernel(const long long* __restrict__ row,
                                const long long* __restrict__ col,
                                const float* __restrict__ dinv,
                                const float* __restrict__ x,
                                float* __restrict__ s) {
  int i = blockIdx.x * blockDim.x + threadIdx.x;
  if (i < NE) {
    int r = (int)row[i];
    atomicAdd(&s[(int)col[i]], dinv[r] * x[r]);
  }
}

// One wave per 16-node tile. Fuses layer-1 (bias+relu) into A-fragment build,
// then h[16x16] @ W2pad[16x16] via 4 chained V_WMMA_F32_16X16X4_F32 (exact f32).
//
// A 16x4 f32 layout: lane l -> row (l&15); VGPR v -> K = 2*(l>=16) + v (per chunk).
// B 4x16  f32 layout: lane l -> col (l&15); VGPR v -> K = 2*(l>=16) + v.
// C/D 16x16 f32: VGPR r, lanes 0-15 -> M=r, lanes 16-31 -> M=r+8; N = l&15.
__global__ void node_transform_kernel(const float* __restrict__ s,
                                      const float* __restrict__ dinv,
                                      const float* __restrict__ x,
                                      const float* __restrict__ W1,
                                      const float* __restrict__ b1,
                                      const float* __restrict__ W2,
                                      float* __restrict__ g,
                                      float* __restrict__ mvec) {
  const int NT = NN / 16;  // 6250 tiles (exact)
  int wave = (blockIdx.x * blockDim.x + threadIdx.x) >> 5;
  int lane = threadIdx.x & 31;
  if (wave >= NT) return;  // wave-uniform: EXEC stays all-1s for WMMA

  int m = lane & 15;
  int node = wave * 16 + m;
  float dv = dinv[node];
  float a = dv * s[node] + dv * dv * x[node];  // layer-1 pre-activation scalar
  int khalf = (lane >> 4) * 2;
  int n = lane & 15;

  v8f acc = {0.f, 0.f, 0.f, 0.f, 0.f, 0.f, 0.f, 0.f};
#pragma unroll
  for (int c = 0; c < 4; ++c) {
    int k0 = 4 * c + khalf;
    int k1 = k0 + 1;
    // fused layer-1: h[m,k] = relu(a*W1[k] + b1[k])
    float h0 = fmaxf(fmaf(a, W1[k0], b1[k0]), 0.0f);
    float h1 = fmaxf(fmaf(a, W1[k1], b1[k1]), 0.0f);
    v2f A = {h0, h1};
    // W2 padded to 16 columns with zeros
    float w0 = (n < F2) ? W2[k0 * F2 + n] : 0.0f;
    float w1 = (n < F2) ? W2[k1 * F2 + n] : 0.0f;
    v2f B = {w0, w1};
    acc = __builtin_amdgcn_wmma_f32_16x16x4_f32(false, A, false, B,
                                                (short)0, acc, false, false);
  }

  if (n < F2) {
    int nodeBase = wave * 16 + ((lane >> 4) ? 8 : 0);
#pragma unroll
    for (int r = 0; r < 8; ++r) {
      int nd = nodeBase + r;
      float val = acc[r];
      g[nd * F2 + n] = val;                 // h @ W2 (for self-loop term)
      mvec[nd * F2 + n] = dinv[nd] * val;   // per-source normalized message
    }
  }
}

__global__ void scatter2_kernel(const long long* __restrict__ row,
                                const long long* __restrict__ col,
                                const float* __restrict__ mvec,
                                float* __restrict__ t) {
  int i = blockIdx.x * blockDim.x + threadIdx.x;
  if (i < NE) {
    int r = (int)row[i];
    int c = (int)col[i];
#pragma unroll
    for (int j = 0; j < F2; ++j)
      atomicAdd(&t[c * F2 + j], mvec[r * F2 + j]);
  }
}

__global__ void finalize_kernel(const float* __restrict__ t,
                                const float* __restrict__ g,
                                const float* __restrict__ dinv,
                                const float* __restrict__ b2,
                                float* __restrict__ out) {
  int i = blockIdx.x * blockDim.x + threadIdx.x;
  if (i < NN) {
    float dv = dinv[i], dv2 = dv * dv;
    float o[F2];
    float mx = -3.4e38f;
#pragma unroll
    for (int j = 0; j < F2; ++j) {
      o[j] = fmaf(dv, t[i * F2 + j], fmaf(dv2, g[i * F2 + j], b2[j]));
      mx = fmaxf(mx, o[j]);
    }
    float sum = 0.0f;
#pragma unroll
    for (int j = 0; j < F2; ++j) sum += expf(o[j] - mx);
    float l = mx + logf(sum);
#pragma unroll
    for (int j = 0; j < F2; ++j) out[i * F2 + j] = o[j] - l;
  }
}

extern "C" void kernel_launch(void* const* d_in, const int* in_sizes, int n_in,
                              void* d_out, int out_size, void* d_ws, size_t ws_size,
                              hipStream_t stream) {
  const float*     x  = (const float*)d_in[0];      // [N,1]
  const long long* ei = (const long long*)d_in[1];  // [2,E] int64
  const float*     W1 = (const float*)d_in[2];      // [1,16]
  const float*     b1 = (const float*)d_in[3];      // [16]
  const float*     W2 = (const float*)d_in[4];      // [16,10]
  const float*     b2 = (const float*)d_in[5];      // [10]
  float* out = (float*)d_out;                        // [N,10]

  const long long* row = ei;
  const long long* col = ei + NE;

  // Workspace layout (floats). Zero-initialized region first (deg, s, t).
  float* ws   = (float*)d_ws;
  float* deg  = ws;             // [N]
  float* s    = ws + NN;        // [N]
  float* t    = ws + 2 * NN;    // [N*10]
  float* dinv = ws + 12 * NN;   // [N]
  float* g    = ws + 13 * NN;   // [N*10]
  float* mvec = ws + 23 * NN;   // [N*10]   total 33*N floats = 13.2 MB

  const int TB = 256;
  int zb  = (12 * NN + TB - 1) / TB;
  int eb  = (NE + TB - 1) / TB;
  int nb  = (NN + TB - 1) / TB;
  int tfb = ((NN / 16) * 32 + TB - 1) / TB;  // 6250 waves * 32 lanes

  zero_kernel<<<zb, TB, 0, stream>>>(ws, 12 * NN);
  deg_kernel<<<eb, TB, 0, stream>>>(col, deg);
  dinv_kernel<<<nb, TB, 0, stream>>>(deg, dinv);
  scatter1_kernel<<<eb, TB, 0, stream>>>(row, col, dinv, x, s);
  node_transform_kernel<<<tfb, TB, 0, stream>>>(s, dinv, x, W1, b1, W2, g, mvec);
  scatter2_kernel<<<eb, TB, 0, stream>>>(row, col, mvec, t);
  finalize_kernel<<<nb, TB, 0, stream>>>(t, g, dinv, b2, out);
}